// GraphSAGE_91216515432812
// MI455X (gfx1250) — compile-verified
//
#include <hip/hip_runtime.h>
#include <cmath>

#define NN 50000
#define EE 600000
#define FD 128
#define CD 40

typedef __attribute__((ext_vector_type(2))) float v2f;
typedef __attribute__((ext_vector_type(8))) float v8f;

// ---------------------------------------------------------------- zero
__global__ void k_zero(float* __restrict__ p, int n4) {
  int i = blockIdx.x * blockDim.x + threadIdx.x;
  if (i < n4) ((float4*)p)[i] = make_float4(0.f, 0.f, 0.f, 0.f);
}

// ---------------------------------------------------------------- degree
__global__ void k_deg(const long long* __restrict__ dst, float* __restrict__ deg) {
  int e = blockIdx.x * blockDim.x + threadIdx.x;
  if (e < EE) atomicAdd(deg + (int)dst[e], 1.0f);
}

__global__ void k_rdeg(float* __restrict__ deg) {
  int i = blockIdx.x * blockDim.x + threadIdx.x;
  if (i < NN) deg[i] = 1.0f / fmaxf(deg[i], 1.0f);
}

// ---------------------------------------------------------------- edge scatter
// one wave (32 lanes) per edge; lane covers 4 consecutive features (float4)
__global__ __launch_bounds__(256) void k_scatter(const float* __restrict__ feat,
                                                 const long long* __restrict__ src,
                                                 const long long* __restrict__ dst,
                                                 float* __restrict__ agg) {
  int t = blockIdx.x * blockDim.x + threadIdx.x;
  int e = t >> 5;
  int lane = t & 31;
  if (e >= EE) return;
  int s = (int)src[e];
  int d = (int)dst[e];
  const float4 v = *(const float4*)(feat + (size_t)s * FD + lane * 4);
  float* a = agg + (size_t)d * FD + lane * 4;
  atomicAdd(a + 0, v.x);
  atomicAdd(a + 1, v.y);
  atomicAdd(a + 2, v.z);
  atomicAdd(a + 3, v.w);
}

// ---------------------------------------------------------------- fused SAGE GEMM
// OUT = relu( [HAS_AGG: (AGG * rdeg) @ Wl +] bl + X @ Wr )
// block = 8 waves; block owns 16 rows, wave w owns cols [16w, 16w+16)
// V_WMMA_F32_16X16X4_F32 fragment layouts per ISA 7.12.2:
//   A 16x4 : lanes 0-15 -> (K=kb+0, kb+1), lanes 16-31 -> (K=kb+2, kb+3), M = lane&15
//   B 4x16 : mirrored, N = lane&15
//   C/D    : vgpr v -> row (v + 8*(lane>>4)), col = lane&15
template <bool HAS_AGG>
__global__ __launch_bounds__(256) void k_gemm(const float* __restrict__ AGG,
                                              const float* __restrict__ X,
                                              const float* __restrict__ rdeg,
                                              const float* __restrict__ Wl,
                                              const float* __restrict__ bl,
                                              const float* __restrict__ Wr,
                                              float* __restrict__ OUT) {
  const int wave = threadIdx.x >> 5;
  const int lane = threadIdx.x & 31;
  const int half = lane >> 4;
  const int l16  = lane & 15;
  const int mBase = blockIdx.x * 16;
  const int nBase = wave * 16;
  const int nCol  = nBase + l16;

  // prefetch weights into cache (64 KB each; 256 threads x 64 floats covers it)
  __builtin_prefetch(Wr + threadIdx.x * 64, 0, 1);
  if (HAS_AGG) __builtin_prefetch(Wl + threadIdx.x * 64, 0, 1);

  v8f acc = {};

  const float* __restrict__ xrow = X + (size_t)(mBase + l16) * FD;
#pragma unroll
  for (int kb = 0; kb < FD; kb += 4) {
    const int kk = kb + 2 * half;
    v2f a; a.x = xrow[kk];            a.y = xrow[kk + 1];
    v2f b; b.x = Wr[kk * FD + nCol];  b.y = Wr[(kk + 1) * FD + nCol];
    acc = __builtin_amdgcn_wmma_f32_16x16x4_f32(false, a, false, b, (short)0, acc,
                                                false, false);
  }

  if (HAS_AGG) {
    const float rd = rdeg[mBase + l16];
    const float* __restrict__ arow = AGG + (size_t)(mBase + l16) * FD;
#pragma unroll
    for (int kb = 0; kb < FD; kb += 4) {
      const int kk = kb + 2 * half;
      v2f a; a.x = arow[kk] * rd;       a.y = arow[kk + 1] * rd;
      v2f b; b.x = Wl[kk * FD + nCol];  b.y = Wl[(kk + 1) * FD + nCol];
      acc = __builtin_amdgcn_wmma_f32_16x16x4_f32(false, a, false, b, (short)0, acc,
                                                  false, false);
    }
  }

  const float bias = bl[nCol];
#pragma unroll
  for (int v = 0; v < 8; ++v) {
    float o = fmaxf(acc[v] + bias, 0.0f);
    OUT[(size_t)(mBase + v + 8 * half) * FD + nCol] = o;
  }
}

// ---------------------------------------------------------------- lin2 + log_softmax
// one wave per node; lane handles class lane (and lane+32 if lane<8)
__global__ __launch_bounds__(256) void k_lin2_lsm(const float* __restrict__ X,
                                                  const float* __restrict__ W2,
                                                  const float* __restrict__ b2,
                                                  float* __restrict__ out) {
  __shared__ float sx[8][FD];
  const int wave = threadIdx.x >> 5;
  const int lane = threadIdx.x & 31;
  const int node = blockIdx.x * 8 + wave;  // N = 6250 * 8, always valid

  const float4 v = *(const float4*)(X + (size_t)node * FD + lane * 4);
  *(float4*)(&sx[wave][lane * 4]) = v;
  __syncthreads();

  const int c0 = lane;
  const int c1 = lane + 32;
  const bool has1 = (c1 < CD);
  const int c1s = has1 ? c1 : 0;
  float a0 = b2[c0];
  float a1 = has1 ? b2[c1] : 0.0f;
#pragma unroll 8
  for (int k = 0; k < FD; ++k) {
    const float xv = sx[wave][k];
    a0 += xv * W2[k * CD + c0];
    a1 += xv * W2[k * CD + c1s];
  }

  float m = fmaxf(a0, has1 ? a1 : -INFINITY);
#pragma unroll
  for (int off = 16; off; off >>= 1) m = fmaxf(m, __shfl_xor(m, off, 32));
  float s = expf(a0 - m) + (has1 ? expf(a1 - m) : 0.0f);
#pragma unroll
  for (int off = 16; off; off >>= 1) s += __shfl_xor(s, off, 32);
  const float lse = m + logf(s);

  out[(size_t)node * CD + c0] = a0 - lse;
  if (has1) out[(size_t)node * CD + c1] = a1 - lse;
}

// ---------------------------------------------------------------- launch
extern "C" void kernel_launch(void* const* d_in, const int* in_sizes, int n_in,
                              void* d_out, int out_size, void* d_ws, size_t ws_size,
                              hipStream_t stream) {
  (void)in_sizes; (void)n_in; (void)out_size; (void)ws_size;

  const float* x        = (const float*)d_in[0];
  const long long* ei   = (const long long*)d_in[1];
  const long long* src  = ei;
  const long long* dst  = ei + EE;
  const float* Wl1 = (const float*)d_in[2];
  const float* bl1 = (const float*)d_in[3];
  const float* Wr1 = (const float*)d_in[4];
  const float* Wl2 = (const float*)d_in[5];
  const float* bl2 = (const float*)d_in[6];
  const float* Wr2 = (const float*)d_in[7];
  const float* Wl3 = (const float*)d_in[8];
  const float* bl3 = (const float*)d_in[9];
  const float* Wr3 = (const float*)d_in[10];
  const float* Wlin1 = (const float*)d_in[11];
  const float* blin1 = (const float*)d_in[12];
  const float* Wlin2 = (const float*)d_in[13];
  const float* blin2 = (const float*)d_in[14];
  float* out = (float*)d_out;

  // workspace layout: agg | buf0 | buf1 | rdeg  (~77 MB)
  float* agg  = (float*)d_ws;
  float* buf0 = agg  + (size_t)NN * FD;
  float* buf1 = buf0 + (size_t)NN * FD;
  float* rdeg = buf1 + (size_t)NN * FD;

  const int feat4 = NN * FD / 4;
  const int zeroFeatB = (feat4 + 255) / 256;
  const int scatterB  = (EE * 32 + 255) / 256;  // one wave per edge
  const int gemmB     = NN / 16;                // 3125
  dim3 b256(256);

  // degrees (shared across layers)
  k_zero<<<(NN / 4 + 255) / 256, b256, 0, stream>>>(rdeg, NN / 4);
  k_deg<<<(EE + 255) / 256, b256, 0, stream>>>(dst, rdeg);
  k_rdeg<<<(NN + 255) / 256, b256, 0, stream>>>(rdeg);

  // layer 1: x -> buf0
  k_zero<<<zeroFeatB, b256, 0, stream>>>(agg, feat4);
  k_scatter<<<scatterB, b256, 0, stream>>>(x, src, dst, agg);
  k_gemm<true><<<gemmB, b256, 0, stream>>>(agg, x, rdeg, Wl1, bl1, Wr1, buf0);

  // layer 2: buf0 -> buf1
  k_zero<<<zeroFeatB, b256, 0, stream>>>(agg, feat4);
  k_scatter<<<scatterB, b256, 0, stream>>>(buf0, src, dst, agg);
  k_gemm<true><<<gemmB, b256, 0, stream>>>(agg, buf0, rdeg, Wl2, bl2, Wr2, buf1);

  // layer 3: buf1 -> buf0
  k_zero<<<zeroFeatB, b256, 0, stream>>>(agg, feat4);
  k_scatter<<<scatterB, b256, 0, stream>>>(buf1, src, dst, agg);
  k_gemm<true><<<gemmB, b256, 0, stream>>>(agg, buf1, rdeg, Wl3, bl3, Wr3, buf0);

  // lin1 (+ReLU): buf0 -> buf1   (X @ Wr branch only)
  k_gemm<false><<<gemmB, b256, 0, stream>>>(agg, buf0, rdeg, Wlin1, blin1, Wlin1, buf1);

  // lin2 + log_softmax: buf1 -> out
  k_lin2_lsm<<<NN / 8, b256, 0, stream>>>(buf1, Wlin2, blin2, out);
}